// GptOssTopKRouter_18923625906264
// MI455X (gfx1250) — compile-verified
//
#include <hip/hip_runtime.h>
#include <hip/hip_bf16.h>
#include <math.h>

#define HIDDEN 2880
#define NEXP   32
#define TOPK   4
#define WAVES_PER_BLOCK 4
#define TOK_PER_TILE    16
#define KSPLIT (HIDDEN / WAVES_PER_BLOCK)   // 720 K-values per wave

typedef __attribute__((ext_vector_type(2))) float v2f;
typedef __attribute__((ext_vector_type(8))) float v8f;

// One workgroup = 4 waves = one 16-token tile.
// Each wave computes a partial 16x32 logit tile over 720 of the 2880 K values
// using V_WMMA_F32_16X16X4_F32, partials are reduced in LDS, then 16 lanes do
// top-4 + softmax + scatter.
__global__ __launch_bounds__(128) void gptoss_topk_router_kernel(
    const float* __restrict__ hs,      // (T, 2880)
    const float* __restrict__ wgt,     // (32, 2880)
    const float* __restrict__ bias,    // (32,)
    float* __restrict__ out_scores,    // (T, 32)
    int*   __restrict__ out_idx,       // (T, 4)
    int T)
{
    __shared__ float lds_part[WAVES_PER_BLOCK * TOK_PER_TILE * NEXP]; // 8 KB
    __shared__ float lds_log[TOK_PER_TILE * NEXP];                    // 2 KB

    const int tid  = threadIdx.x;
    const int wave = tid >> 5;
    const int lane = tid & 31;
    const int half = lane >> 4;   // 0: lanes 0-15, 1: lanes 16-31
    const int l15  = lane & 15;
    const int tokBase = blockIdx.x * TOK_PER_TILE;

    // A operand: fp32 16x4, lane<16 holds (k,k+1) of row M=lane,
    //            lane>=16 holds (k+2,k+3) of row M=lane-16.
    // B operand (weight^T, 4x16): mirrored layout over columns N.
    const float* aRow  = hs  + (size_t)(tokBase + l15) * HIDDEN;
    const float* b0Row = wgt + (size_t)l15        * HIDDEN;   // experts 0-15
    const float* b1Row = wgt + (size_t)(l15 + 16) * HIDDEN;   // experts 16-31

    const int kBeg = wave * KSPLIT;
    const int koff = 2 * half;

    // Per-lane base pointers for this wave's K slice; inner loop then uses
    // pure immediate offsets (no per-iteration 64-bit address math).
    const float* aP  = aRow  + kBeg + koff;
    const float* b0P = b0Row + kBeg + koff;
    const float* b1P = b1Row + kBeg + koff;

    // Warm the first cachelines of the streams (one-shot, outside the loop).
    __builtin_prefetch(aP, 0, 0);
    __builtin_prefetch(aP + 32, 0, 0);

    v8f acc0 = {}; // experts 0-15
    v8f acc1 = {}; // experts 16-31

    #pragma unroll 4
    for (int k = 0; k < KSPLIT; k += 4) {
        v2f a  = *(const v2f*)(aP  + k);
        v2f b0 = *(const v2f*)(b0P + k);
        v2f b1 = *(const v2f*)(b1P + k);
        // D = A(16x4) * B(4x16) + C   (8 args: neg_a,A,neg_b,B,c_mod,C,reuse_a,reuse_b)
        acc0 = __builtin_amdgcn_wmma_f32_16x16x4_f32(false, a, false, b0,
                                                     (short)0, acc0, false, false);
        acc1 = __builtin_amdgcn_wmma_f32_16x16x4_f32(false, a, false, b1,
                                                     (short)0, acc1, false, false);
    }

    // Stash partial tiles: C/D layout -> VGPR r: lanes 0-15 = M=r, lanes 16-31 = M=r+8.
    {
        float* base = lds_part + wave * (TOK_PER_TILE * NEXP);
        #pragma unroll
        for (int r = 0; r < 8; ++r) {
            const int m = r + 8 * half;
            base[m * NEXP + l15]      = acc0[r];
            base[m * NEXP + 16 + l15] = acc1[r];
        }
    }
    __syncthreads();

    // Reduce the 4 K-slice partials + bias into final logits (512 entries, 4/thread).
    #pragma unroll
    for (int i = 0; i < 4; ++i) {
        const int e = tid * 4 + i;            // e = m*32 + n
        const int n = e & (NEXP - 1);
        float s = bias[n];
        #pragma unroll
        for (int wv = 0; wv < WAVES_PER_BLOCK; ++wv)
            s += lds_part[wv * (TOK_PER_TILE * NEXP) + e];
        lds_log[e] = s;
    }
    __syncthreads();

    // Top-4 + softmax + scatter: one lane per token (trivial vs. the GEMM).
    if (tid < TOK_PER_TILE) {
        float* row = lds_log + tid * NEXP;
        float vals[TOPK]; int idxs[TOPK];
        #pragma unroll
        for (int j = 0; j < TOPK; ++j) {
            float best = -INFINITY; int bi = 0;
            for (int e = 0; e < NEXP; ++e) {
                const float v = row[e];
                if (v > best) { best = v; bi = e; }   // strict > => lowest index on ties
            }
            vals[j] = best; idxs[j] = bi;
            row[bi] = -INFINITY;
        }
        float p[TOPK]; float s = 0.f;
        #pragma unroll
        for (int j = 0; j < TOPK; ++j) { p[j] = __expf(vals[j] - vals[0]); s += p[j]; }
        const float inv = 1.f / s;

        const int tok = tokBase + tid;
        float* orow = out_scores + (size_t)tok * NEXP;
        for (int e = 0; e < NEXP; ++e) {
            float v = 0.f;
            #pragma unroll
            for (int j = 0; j < TOPK; ++j) if (e == idxs[j]) v = p[j] * inv;
            orow[e] = v;
        }
        #pragma unroll
        for (int j = 0; j < TOPK; ++j)
            out_idx[(size_t)tok * TOPK + j] = idxs[j];
    }
}

extern "C" void kernel_launch(void* const* d_in, const int* in_sizes, int n_in,
                              void* d_out, int out_size, void* d_ws, size_t ws_size,
                              hipStream_t stream) {
    const float* hs   = (const float*)d_in[0];   // (4,2048,2880) f32
    const float* wgt  = (const float*)d_in[1];   // (32,2880) f32
    const float* bias = (const float*)d_in[2];   // (32,) f32

    const int T = in_sizes[0] / HIDDEN;          // 8192 tokens

    float* out_scores = (float*)d_out;                            // T*32 floats
    int*   out_idx    = (int*)((float*)d_out + (size_t)T * NEXP); // then T*4 int32

    const int grid = T / TOK_PER_TILE;           // 512 blocks, 4 waves each
    gptoss_topk_router_kernel<<<grid, WAVES_PER_BLOCK * 32, 0, stream>>>(
        hs, wgt, bias, out_scores, out_idx, T);
}